// ConcatAttention_27419071217839
// MI455X (gfx1250) — compile-verified
//
#include <hip/hip_runtime.h>
#include <math.h>

typedef __attribute__((ext_vector_type(2))) float v2f;
typedef __attribute__((ext_vector_type(4))) float v4f;
typedef __attribute__((ext_vector_type(8))) float v8f;

#define NEG_INF_F (-1000000.0f)
#define PITCH 260   // 256 + 4 floats pad: banks (4*row + col) % 64 -> conflict-free, keeps 16B align

// ---------------------------------------------------------------------------
// K0: targetT[b,d] = sum_q input[b,q] * W_q[d,q]        (32 x 1024)
// ---------------------------------------------------------------------------
__global__ void __launch_bounds__(256)
k_target(const float* __restrict__ inp, const float* __restrict__ Wq,
         float* __restrict__ tT) {
  int b = blockIdx.y;
  int d = blockIdx.x * 256 + threadIdx.x;
  const v4f* wr = (const v4f*)(Wq + (size_t)d * 1024);
  const v4f* ir = (const v4f*)(inp + (size_t)b * 1024);
  float acc = 0.f;
#pragma unroll 8
  for (int q = 0; q < 256; ++q) {
    v4f w = wr[q], x = ir[q];
    acc += w.x * x.x + w.y * x.y + w.z * x.z + w.w * x.w;
  }
  tT[b * 1024 + d] = acc;
}

// ---------------------------------------------------------------------------
// K1: precompute = context @ W_pre^T + b_pre   (65536 x 1024, f32 WMMA)
//     fused: energy[b,s] = sum_d tanh(precompute + targetT[b,d]) * v[d]
// WG = 128 threads (4 waves), owns 64 rows; wave w owns rows [w*16, w*16+16).
// Context staged to LDS with double-buffered GLOBAL_LOAD_ASYNC_TO_LDS_B128
// (ASYNCcnt), overlapping next-chunk staging with WMMA on current chunk.
// ---------------------------------------------------------------------------

// issue one 64x256 chunk (32 async b128 loads per thread; 128 threads)
__device__ __forceinline__ void stage_async(const float* ctx_blk, int kc,
                                            float* ldsbuf, int tid) {
#pragma unroll
  for (int i = 0; i < 32; ++i) {
    int idx = tid + i * 128;
    int row = idx >> 6;
    int c4  = (idx & 63) << 2;
    const float* g = ctx_blk + (size_t)row * 1024 + kc * 256 + c4;
    unsigned lds_off = (unsigned)(size_t)(ldsbuf + row * PITCH + c4);
    asm volatile("global_load_async_to_lds_b128 %0, %1, off"
                 :: "v"(lds_off), "v"((unsigned long long)(size_t)g)
                 : "memory");
  }
}

__global__ void __launch_bounds__(128)
k_gemm_energy(const float* __restrict__ ctx, const float* __restrict__ Wp,
              const float* __restrict__ bpre, const float* __restrict__ vvec,
              const float* __restrict__ tT,
              float* __restrict__ pre_out, float* __restrict__ energy_out) {
  __shared__ float Alds[2][64 * PITCH];   // double-buffered 64x256 chunks
  __shared__ float elds[64];

  const int tid  = threadIdx.x;
  const int wave = tid >> 5;
  const int lane = tid & 31;
  const int m16  = lane & 15;      // row-in-tile / col-in-tile selector
  const int hi   = lane >> 4;      // half-wave selector
  const int R0   = blockIdx.x * 64;     // flattened (b*2048 + s) row base
  const int b    = R0 >> 11;
  const float* ctx_blk = ctx + (size_t)R0 * 1024;

  if (tid < 64) elds[tid] = 0.f;

  float e[8] = {0.f, 0.f, 0.f, 0.f, 0.f, 0.f, 0.f, 0.f};
  v8f acc[8];

  // prologue: kick off chunk 0 into buffer 0
  stage_async(ctx_blk, 0, Alds[0], tid);

  // flat pipeline over 32 chunk-uses: ng = c>>2 (n-group), kc = c&3 (K chunk)
  for (int c = 0; c < 32; ++c) {
    const int kc  = c & 3;
    const int ng  = c >> 2;
    const int cur = c & 1;

    if (kc == 0) {
#pragma unroll
      for (int t = 0; t < 8; ++t)
        acc[t] = (v8f){0.f, 0.f, 0.f, 0.f, 0.f, 0.f, 0.f, 0.f};
    }

    // barrier A: everyone finished computing on the buffer we are about to
    // overwrite (used at iteration c-1), so it is safe to re-fill it.
    __syncthreads();
    if (c + 1 < 32) {
      stage_async(ctx_blk, (c + 1) & 3, Alds[(c + 1) & 1], tid);
      // 64 loads in flight; wait until only next-chunk's 32 remain
      asm volatile("s_wait_asynccnt 0x20" ::: "memory");
    } else {
      asm volatile("s_wait_asynccnt 0x0" ::: "memory");
    }
    // barrier B: all waves' async writes for chunk c have landed in LDS.
    __syncthreads();

    // A frag base: 16x4 f32 layout -> lanes 0-15: K=k..k+1, lanes 16-31: K=k+2..k+3
    const float* abase = &Alds[cur][(wave * 16 + m16) * PITCH + 2 * hi];
#pragma unroll
    for (int t = 0; t < 8; ++t) {
      const int dcol = ng * 128 + t * 16 + m16;   // B col (N) for this lane
      const float* bbase = Wp + (size_t)dcol * 1024 + kc * 256 + 2 * hi;
#pragma unroll 8
      for (int k = 0; k < 256; k += 4) {
        v2f av = *(const v2f*)(abase + k);   // ds_load_b64
        v2f bv = *(const v2f*)(bbase + k);   // global_load_b64 (L2-hot W_pre)
        acc[t] = __builtin_amdgcn_wmma_f32_16x16x4_f32(
            false, av, false, bv, (short)0, acc[t], false, false);
      }
    }

    // epilogue once per n-group: +b_pre, store precompute, tanh-dot energy
    if (kc == 3) {
      const int rowbase = R0 + wave * 16 + 8 * hi;  // C layout: vgpr r -> M = r + 8*hi
#pragma unroll
      for (int t = 0; t < 8; ++t) {
        const int n  = ng * 128 + t * 16 + m16;
        const float bp = bpre[n];
        const float tt = tT[b * 1024 + n];
        const float vn = vvec[n];
#pragma unroll
        for (int r = 0; r < 8; ++r) {
          float p = acc[t][r] + bp;
          pre_out[(size_t)(rowbase + r) * 1024 + n] = p;
          e[r] += tanhf(p + tt) * vn;
        }
      }
    }
  }

  // reduce per-lane energy partials across the wave into per-row sums
#pragma unroll
  for (int r = 0; r < 8; ++r)
    atomicAdd(&elds[wave * 16 + 8 * hi + r], e[r]);
  __syncthreads();
  if (tid < 64) energy_out[R0 + tid] = elds[tid];
}

// ---------------------------------------------------------------------------
// K2: masked softmax over S per batch
// ---------------------------------------------------------------------------
__global__ void __launch_bounds__(256)
k_softmax(const float* __restrict__ energy, const int* __restrict__ mask,
          float* __restrict__ score) {
  __shared__ float red[256];
  const int b = blockIdx.x, tid = threadIdx.x;
  float le[8];
  float mx = -3.4e38f;
#pragma unroll
  for (int i = 0; i < 8; ++i) {
    int s = tid + i * 256;
    float v = energy[b * 2048 + s];
    if (mask[b * 2048 + s] == 0) v = NEG_INF_F;
    le[i] = v;
    mx = fmaxf(mx, v);
  }
  red[tid] = mx;
  __syncthreads();
  for (int off = 128; off > 0; off >>= 1) {
    if (tid < off) red[tid] = fmaxf(red[tid], red[tid + off]);
    __syncthreads();
  }
  mx = red[0];
  __syncthreads();
  float sum = 0.f;
#pragma unroll
  for (int i = 0; i < 8; ++i) {
    le[i] = expf(le[i] - mx);
    sum += le[i];
  }
  red[tid] = sum;
  __syncthreads();
  for (int off = 128; off > 0; off >>= 1) {
    if (tid < off) red[tid] += red[tid + off];
    __syncthreads();
  }
  float inv = 1.f / red[0];
#pragma unroll
  for (int i = 0; i < 8; ++i)
    score[b * 2048 + tid + i * 256] = le[i] * inv;
}

// ---------------------------------------------------------------------------
// K3: weightedContext[b,a] = sum_s score[b,s] * context[b,s,a]
// ---------------------------------------------------------------------------
__global__ void __launch_bounds__(256)
k_weighted(const float* __restrict__ ctx, const float* __restrict__ score,
           float* __restrict__ wc) {
  __shared__ float sc[2048];
  const int b = blockIdx.y, tid = threadIdx.x;
#pragma unroll
  for (int i = 0; i < 8; ++i)
    sc[tid + i * 256] = score[b * 2048 + tid + i * 256];
  __syncthreads();
  const int a = blockIdx.x * 256 + tid;
  const float* cb = ctx + (size_t)b * 2048 * 1024 + a;
  float acc = 0.f;
#pragma unroll 8
  for (int s = 0; s < 2048; ++s)
    acc += sc[s] * cb[(size_t)s * 1024];
  wc[b * 1024 + a] = acc;
}

// ---------------------------------------------------------------------------
extern "C" void kernel_launch(void* const* d_in, const int* in_sizes, int n_in,
                              void* d_out, int out_size, void* d_ws, size_t ws_size,
                              hipStream_t stream) {
  const float* inp  = (const float*)d_in[0];   // (32,1,1024)
  const float* ctx  = (const float*)d_in[1];   // (32,2048,1024)
  const float* Wpre = (const float*)d_in[2];   // (1024,1024)
  const float* bpre = (const float*)d_in[3];   // (1024,)
  const float* Wq   = (const float*)d_in[4];   // (1024,1024)
  const float* vv   = (const float*)d_in[5];   // (1024,)
  const int*   mask = (const int*)d_in[6];     // (32,1,2048)

  float* out    = (float*)d_out;
  float* wc     = out;                 // 32*1024
  float* score  = out + 32768;         // 32*2048
  float* pre    = out + 98304;         // 32*2048*1024

  float* ws     = (float*)d_ws;
  float* tT     = ws;                  // 32*1024
  float* energy = ws + 32768;          // 32*2048

  k_target<<<dim3(4, 32), 256, 0, stream>>>(inp, Wq, tT);
  k_gemm_energy<<<dim3(1024), 128, 0, stream>>>(ctx, Wpre, bpre, vv, tT, pre, energy);
  k_softmax<<<dim3(32), 256, 0, stream>>>(energy, mask, score);
  k_weighted<<<dim3(4, 32), 256, 0, stream>>>(ctx, score, wc);
}